// ArnoldCat_23536420782184
// MI455X (gfx1250) — compile-verified
//
#include <hip/hip_runtime.h>

// Arnold cat map, 7 iterations composed into one gather:
//   M = [[1,1],[2,1]],  M^7 mod 1024 = [[239,169],[338,239]]  (det = -1)
//   out[batch, a, b, c] = in[batch, (239a+169b)&1023, (338a+239b)&1023, c]
//
// Layout (NHWC): flat = ((batch*1024 + row)*1024 + col)*3 + c, float32.
// 16 images x 12 MB each; per-image working set fits easily in 192 MB L2,
// so the scattered 12 B reads hit L2 and HBM only streams 192 MiB in +
// 192 MiB out (~17 us at 23.3 TB/s). Stores are non-temporal to keep L2
// capacity dedicated to the gather side.

#define LOG_N 10
#define N_DIM 1024u

struct __attribute__((packed, aligned(4))) Pix3 { float x, y, z; };

typedef __attribute__((ext_vector_type(4))) float v4f;   // native vector: OK for NT builtin

__global__ __launch_bounds__(256) void arnold7_gather(const float* __restrict__ in,
                                                      float* __restrict__ out,
                                                      unsigned int nRows) {
    // One block == one output row of one batch image (256 threads * 4 pixels).
    const unsigned int row = blockIdx.x;          // 0 .. 16*1024-1
    if (row >= nRows) return;

    const unsigned int batch = row >> LOG_N;      // SALU (uniform per block)
    const unsigned int a     = row & (N_DIM - 1u);
    const unsigned int base  = batch << (2 * LOG_N);   // pixel index of image start
    const unsigned int ra    = 239u * a;          // row term   (scalar)
    const unsigned int ca    = 338u * a;          // col term   (scalar)

    const unsigned int b0 = threadIdx.x << 2;     // 4 consecutive output cols

    float v[12];
#pragma unroll
    for (int i = 0; i < 4; ++i) {
        const unsigned int b  = b0 + (unsigned int)i;
        const unsigned int sa = (ra + 169u * b) & (N_DIM - 1u);
        const unsigned int sb = (ca + 239u * b) & (N_DIM - 1u);
        const unsigned int srcPix = base + (sa << LOG_N) + sb;
        // 12-byte gather (global_load_b96), 4-byte aligned; L2-resident.
        const Pix3 p = *(const Pix3*)(in + (size_t)srcPix * 3u);
        v[i * 3 + 0] = p.x;
        v[i * 3 + 1] = p.y;
        v[i * 3 + 2] = p.z;
    }

    // Destination: 4 pixels = 48 bytes, 16-byte aligned, fully coalesced.
    const size_t dst = ((size_t)row * N_DIM + b0) * 3u;
    v4f* o = (v4f*)(out + dst);
    v4f s0 = { v[0], v[1], v[2],  v[3]  };
    v4f s1 = { v[4], v[5], v[6],  v[7]  };
    v4f s2 = { v[8], v[9], v[10], v[11] };
    __builtin_nontemporal_store(s0, o + 0);
    __builtin_nontemporal_store(s1, o + 1);
    __builtin_nontemporal_store(s2, o + 2);
}

extern "C" void kernel_launch(void* const* d_in, const int* in_sizes, int n_in,
                              void* d_out, int out_size, void* d_ws, size_t ws_size,
                              hipStream_t stream) {
    const float* in  = (const float*)d_in[0];
    float*       out = (float*)d_out;

    // out_size = 16*1024*1024*3 floats -> one block per output row.
    const unsigned int nRows = (unsigned int)(out_size / (int)(3u * N_DIM)); // 16384
    dim3 block(256);
    dim3 grid(nRows);
    arnold7_gather<<<grid, block, 0, stream>>>(in, out, nRows);
}